// orthogonal_projection_9637906612751
// MI455X (gfx1250) — compile-verified
//
#include <hip/hip_runtime.h>

// ---------------------------------------------------------------------------
// orthogonal projection: out[n,c] = expm( sum_i dX[n,i] * skew(A)[i,c] )
// N=8192, INPUT=64, C=4, H=32.  One wave32 per (n,c) matrix; expm matmuls via
// V_WMMA_F32_16X16X4_F32 chained through LDS.
// ---------------------------------------------------------------------------

typedef __attribute__((ext_vector_type(2))) float v2f;
typedef __attribute__((ext_vector_type(8))) float v8f;

#define N_TOT     8192
#define INPUT_SZ  64
#define CHANNELS  4
#define HID       32
#define LDST      33                       // padded LDS row stride (floats)
#define MAT_F     (HID * LDST)             // 1056 floats per 32x32 buffer
#define WPB       4                        // waves per block
#define WAVE_LDS  (3 * MAT_F + 64)         // X, H, T + dX row

// 32x32 (row-major, stride LDST) matmul: Cb = Ab @ Bb, via 16x16x4 f32 WMMA.
// Layouts per CDNA5 ISA 7.12.2:
//   A 16x4 : VGPR v, lane = {half,l16}:  A[M=l16][K = k0 + v + 2*half]
//   B 4x16 : VGPR v:                     B[K = k0 + v + 2*half][N=l16]
//   C 16x16: VGPR v:                     C[M = v + 8*half][N=l16]
__device__ __forceinline__ void mm32(const float* Ab, const float* Bb,
                                     float* Cb, int l16, int half) {
#pragma unroll
  for (int mi = 0; mi < 32; mi += 16) {
#pragma unroll
    for (int ni = 0; ni < 32; ni += 16) {
      v8f acc = {0.f, 0.f, 0.f, 0.f, 0.f, 0.f, 0.f, 0.f};
#pragma unroll
      for (int k0 = 0; k0 < 32; k0 += 4) {
        const int ka = k0 + 2 * half;
        v2f a, b;
        a.x = Ab[(mi + l16) * LDST + ka];
        a.y = Ab[(mi + l16) * LDST + ka + 1];
        b.x = Bb[ka * LDST + ni + l16];
        b.y = Bb[(ka + 1) * LDST + ni + l16];
        acc = __builtin_amdgcn_wmma_f32_16x16x4_f32(
            false, a, false, b, (short)0, acc, false, false);
      }
#pragma unroll
      for (int v = 0; v < 8; ++v)
        Cb[(mi + v + 8 * half) * LDST + ni + l16] = acc[v];
    }
  }
}

// S[i,c,h,k] = 0.5 * (A[i,c,h,k] - A[i,c,k,h])
__global__ void __launch_bounds__(256)
skew_kernel(const float* __restrict__ A, float* __restrict__ S) {
  const int idx = blockIdx.x * 256 + threadIdx.x;  // 262144 total
  const int k  = idx & 31;
  const int h  = (idx >> 5) & 31;
  const int ic = idx >> 10;
  const float a  = A[idx];
  const float at = A[(ic << 10) + (k << 5) + h];
  S[idx] = 0.5f * (a - at);
}

__global__ void __launch_bounds__(WPB * 32)
expm_fused_kernel(const float* __restrict__ dX, const float* __restrict__ S,
                  float* __restrict__ out) {
  __shared__ float lds[WPB * WAVE_LDS];

  const int tid  = threadIdx.x;
  const int wave = tid >> 5;
  const int lane = tid & 31;
  const int l16  = lane & 15;
  const int half = lane >> 4;

  float* Xb  = &lds[wave * WAVE_LDS];
  float* Hb  = Xb + MAT_F;
  float* Tb  = Hb + MAT_F;
  float* dxb = Tb + MAT_F;

  const int m = blockIdx.x * WPB + wave;   // matrix id in [0, 32768)
  const int n = m >> 2;
  const int c = m & 3;

  // stage this wave's dX row (64 floats) into LDS
  dxb[lane]      = dX[n * INPUT_SZ + lane];
  dxb[lane + 32] = dX[n * INPUT_SZ + lane + 32];

  // ---- phase 1: lane L accumulates row L of AX = sum_i dX[n,i]*S[i,c] ----
  float ax[32];
#pragma unroll
  for (int j = 0; j < 32; ++j) ax[j] = 0.f;

  const float4* S4 = reinterpret_cast<const float4*>(S);
  for (int i = 0; i < INPUT_SZ; ++i) {
    const float a = dxb[i];  // wave-uniform LDS broadcast
    const float4* row =
        S4 + (size_t)((i * CHANNELS + c) * HID + lane) * 8;
#pragma unroll
    for (int q = 0; q < 8; ++q) {
      const float4 s4 = row[q];
      ax[4 * q + 0] += a * s4.x;
      ax[4 * q + 1] += a * s4.y;
      ax[4 * q + 2] += a * s4.z;
      ax[4 * q + 3] += a * s4.w;
    }
  }

  // ---- inf-norm -> scaling exponent s (wave-uniform) ----
  float rs = 0.f;
#pragma unroll
  for (int j = 0; j < 32; ++j) rs += fabsf(ax[j]);
#pragma unroll
  for (int off = 16; off > 0; off >>= 1)
    rs = fmaxf(rs, __shfl_xor(rs, off, 32));
  int s = 0;
  float sc = 1.f;
  while (rs * sc > 0.5f && s < 30) { sc *= 0.5f; ++s; }

  // ---- stage Xs and Horner seed H = I + Xs/8 into LDS ----
#pragma unroll
  for (int j = 0; j < 32; ++j) {
    const float xs = ax[j] * sc;
    Xb[lane * LDST + j] = xs;
    Hb[lane * LDST + j] = xs * 0.125f + ((j == lane) ? 1.f : 0.f);
  }

  // ---- Taylor-8 Horner: H = I + (Xs * (1/j)) @ H, j = 7..1 ----
  // Unrolled with literal reciprocals: no v_div/v_rcp between matmuls.
  constexpr float kInv[8] = {0.f,          1.f,          0.5f,
                             1.f / 3.f,    0.25f,        0.2f,
                             1.f / 6.f,    1.f / 7.f};
#pragma unroll
  for (int j = 7; j >= 1; --j) {
    mm32(Xb, Hb, Tb, l16, half);
    const float inv = kInv[j];  // compile-time literal after unroll
#pragma unroll
    for (int q = 0; q < 32; ++q)
      Hb[lane * LDST + q] =
          Tb[lane * LDST + q] * inv + ((q == lane) ? 1.f : 0.f);
  }

  // ---- s squarings (uniform count): P <- P @ P ----
  float* Pa = Hb;
  float* Pb = Tb;
  for (int t = 0; t < s; ++t) {
    mm32(Pa, Pa, Pb, l16, half);
    float* tp = Pa; Pa = Pb; Pb = tp;
  }

  // ---- write row L (32 floats) to out[m] ----
  float4* o4 = reinterpret_cast<float4*>(out) + (size_t)m * 256 + lane * 8;
#pragma unroll
  for (int q = 0; q < 8; ++q) {
    float4 r;
    r.x = Pa[lane * LDST + 4 * q + 0];
    r.y = Pa[lane * LDST + 4 * q + 1];
    r.z = Pa[lane * LDST + 4 * q + 2];
    r.w = Pa[lane * LDST + 4 * q + 3];
    o4[q] = r;
  }
}

extern "C" void kernel_launch(void* const* d_in, const int* in_sizes, int n_in,
                              void* d_out, int out_size, void* d_ws,
                              size_t ws_size, hipStream_t stream) {
  const float* dX = (const float*)d_in[0];  // (8192, 64)
  const float* A  = (const float*)d_in[1];  // (64, 4, 32, 32)
  float* out = (float*)d_out;               // (8192, 4, 32, 32)
  float* S   = (float*)d_ws;                // 1 MB scratch for skew(A)

  skew_kernel<<<(INPUT_SZ * CHANNELS * HID * HID) / 256, 256, 0, stream>>>(A, S);
  expm_fused_kernel<<<(N_TOT * CHANNELS) / WPB, WPB * 32, 0, stream>>>(dX, S, out);
}